// ST_GCN_Block_59055800320605
// MI455X (gfx1250) — compile-verified
//
#include <hip/hip_runtime.h>
#include <stdint.h>

// ---------------------------------------------------------------------------
// Fused ST-GCN block for MI455X (gfx1250), fp32 end-to-end via
// V_WMMA_F32_16X16X4_F32. One workgroup = one (batch b, 4-timestep) tile.
//   Z = sum_s W_s @ (X @ PA_s)  + bias/BN/ReLU/residual (all fused in LDS).
// - X tile staged with GLOBAL_LOAD_ASYNC_TO_LDS_B32 (per-lane gather -> LDS,
//   ASYNCcnt path; the M-interleave makes TDM's contiguous-dim rule a no-fit).
// - K-pair-interleaved LDS layouts ([k/2][col][2]) so every B-fragment is one
//   aligned ds_load_b64 and every D writeback is a dense ds_store_b64.
// - LDS arrays ordered hottest-first so DS address constants fold into the
//   16-bit ds_load offset field instead of per-access v_or_b32.
// - K-outer GEMM loops reuse each A-fragment across 4-6 accumulator tiles.
// ---------------------------------------------------------------------------

typedef float v2f __attribute__((ext_vector_type(2)));
typedef float v8f __attribute__((ext_vector_type(8)));

#define BN_EPS 1e-5f

// Problem dims (fixed by the reference)
#define N_  32
#define C_  64
#define T_  300
#define V_  25
#define M_  2
#define S_  3
#define O_  64
#define B_  (N_ * M_)      // 64 fused batch
#define TT  4              // timesteps per workgroup (300 = 75*4 exact)
#define VP  32             // V padded to 32
#define KC  (S_ * C_)      // 192  (K of the channel GEMM)
#define NCOL (TT * VP)     // 128  (N of the channel GEMM)

__device__ __forceinline__ v8f wmma_f32(v2f a, v2f b, v8f c) {
    return __builtin_amdgcn_wmma_f32_16x16x4_f32(false, a, false, b,
                                                 (short)0, c, false, false);
}

__global__ __launch_bounds__(256, 1)
void stgcn_fused_kernel(const float* __restrict__ x,
                        const float* __restrict__ PA,
                        const float* __restrict__ conv_w,
                        const float* __restrict__ conv_b,
                        const float* __restrict__ bn_gamma,
                        const float* __restrict__ bn_beta,
                        const float* __restrict__ bn_mean,
                        const float* __restrict__ bn_var,
                        float* __restrict__ out)
{
    // ---- LDS (~142 KB), hottest arrays first (low offsets -> DS immediates) ----
    // sXP: K-pair interleaved: element (k, n) -> [((k/2)*NCOL + n)*2 + k%2]
    __shared__ __attribute__((aligned(16))) float sXP[KC * NCOL];      // 96 KB @0
    // sPA: K-pair interleaved: element (s, v, w) -> [((s*16 + v/2)*VP + w)*2 + v%2]
    __shared__ __attribute__((aligned(16))) float sPA[S_ * VP * VP];   // 12 KB
    // sX: plain row-major, rows=(c,t), cols=v (padded)
    __shared__ __attribute__((aligned(16))) float sX[C_ * TT * VP];    // 32 KB
    __shared__ float sColsum[S_ * VP];
    __shared__ float sScale[O_];
    __shared__ float sShift[O_];
    __shared__ float sBias[KC];

    const int tidx = (int)threadIdx.x;
    const int lane = tidx & 31;            // wave32
    const int wave = tidx >> 5;            // 0..7
    const int half = lane >> 4;            // 0/1  (K-pair select per ISA layout)
    const int ln   = lane & 15;            // row/col within 16

    const int tb    = (int)blockIdx.x;     // 0..74
    const int b     = (int)blockIdx.y;     // 0..63
    const int Tbase = tb * TT;
    const int n_idx = b >> 1;              // b / M
    const int m_idx = b & 1;               // b % M

    // ================= stage 0: stage constants + X tile =================
    // Warm GL2 with conv_w (48 KB, reused by every K-step of stage 2).
    if (tidx < 192) __builtin_prefetch(conv_w + tidx * 64, 0, 0);

    for (int i = tidx; i < S_ * VP * VP; i += 256) {
        int s = i / (VP * VP);
        int r = i % (VP * VP);
        int v = r / VP, w = r % VP;
        float val = (v < V_ && w < V_) ? PA[(s * V_ + v) * V_ + w] : 0.0f;
        sPA[((s * (VP / 2) + (v >> 1)) * VP + w) * 2 + (v & 1)] = val;
    }
    if (tidx < S_ * VP) {
        int s = tidx / VP, w = tidx % VP;
        float acc = 0.0f;
        if (w < V_)
            for (int v = 0; v < V_; ++v) acc += PA[(s * V_ + v) * V_ + w];
        sColsum[tidx] = acc;
    }
    if (tidx < O_) {
        float sc = bn_gamma[tidx] * rsqrtf(bn_var[tidx] + BN_EPS);
        sScale[tidx] = sc;
        sShift[tidx] = bn_beta[tidx] - bn_mean[tidx] * sc;
    }
    if (tidx < KC) sBias[tidx] = conv_b[tidx];

    // X tile via async gather -> LDS (ASYNCcnt path). x layout (N,C,T,V,M);
    // v-dimension is 8B-strided (M interleave): per-lane global & LDS addresses.
    for (int i = tidx; i < C_ * TT * VP; i += 256) {
        int v = i & 31;
        if (v < V_) {
            int r = i >> 5;                // c*TT + t
            int c = r >> 2, t = r & 3;
            size_t gi = ((((size_t)n_idx * C_ + c) * T_ + (Tbase + t)) * V_ + v) * M_ + m_idx;
            unsigned long long ga = (unsigned long long)(uintptr_t)(x + gi);
            unsigned la = (unsigned)(uintptr_t)(&sX[i]);   // LDS byte address = low 32 bits of pointer
            asm volatile("global_load_async_to_lds_b32 %0, %1, off"
                         :: "v"(la), "v"(ga) : "memory");
        } else {
            sX[i] = 0.0f;                  // pad joints 25..31
        }
    }
    asm volatile("s_wait_asynccnt 0" ::: "memory");
    __syncthreads();

    // ========== stage 1: XP = X @ PA_s   (per-subset joint mixing) ==========
    // Each wave owns two 16-row blocks over (c,t); A-fragment shared across
    // all 6 (s,nt) output tiles -> 6 WMMAs per ds_load_b64.
    for (int p = 0; p < 2; ++p) {
        int mt = wave * 2 + p;             // 0..15
        int Rb = mt * 16;
        v8f acc[6] = {};                   // idx q = s*2 + nt
        #pragma unroll
        for (int kk = 0; kk < VP / 4; ++kk) {
            int k0  = kk * 4 + 2 * half;   // even; lanes0-15: K0,1  lanes16-31: K2,3
            int kp  = k0 >> 1;             // pair index
            v2f a = *(const v2f*)&sX[(Rb + ln) * VP + k0];
            #pragma unroll
            for (int q = 0; q < 6; ++q) {
                int s = q >> 1, nt = q & 1;
                v2f bb = *(const v2f*)&sPA[((s * (VP / 2) + kp) * VP + nt * 16 + ln) * 2];
                acc[q] = wmma_f32(a, bb, acc[q]);
            }
        }
        // D writeback: element i (k-row even) pairs with i+4 (k-row+1, same n)
        // -> one dense ds_store_b64 per pair, conflict-free (8B lane stride).
        #pragma unroll
        for (int q = 0; q < 6; ++q) {
            int s = q >> 1, nt = q & 1;
            #pragma unroll
            for (int i = 0; i < 4; ++i) {
                int rr = Rb + i + 8 * half;    // even-c element (D: VGPR i -> M=i/i+8)
                int c  = rr >> 2, t = rr & 3;
                int kRow = s * C_ + c;         // even
                int n = t * VP + nt * 16 + ln;
                v2f d; d.x = acc[q][i]; d.y = acc[q][i + 4];
                *(v2f*)&sXP[(kRow >> 1) * (NCOL * 2) + n * 2] = d;
            }
        }
    }
    __syncthreads();

    // ========== stage 2: Z = W2 @ XP   (channel mixing, K = 192) ==========
    // W2[o, k=s*64+c] = conv_w[s*64+o, c] read from global (GL2-hot 48 KB).
    // Each wave: one M-tile (o) x four N-tiles; A-fragment reused 4x per K step.
    {
        int Mb  = (wave >> 1) * 16;        // 0,16,32,48
        int ntb = (wave & 1) * 4;          // N-tile base: 0 or 4
        v8f acc[4] = {};
        #pragma unroll 8
        for (int kk = 0; kk < KC / 4; ++kk) {
            int k0 = kk * 4 + 2 * half;    // even -> never crosses a 64-boundary
            int kp = k0 >> 1;
            int s  = k0 >> 6;
            int c  = k0 & 63;
            v2f a = *(const v2f*)(conv_w + ((size_t)(s * O_ + Mb + ln)) * C_ + c);
            #pragma unroll
            for (int j = 0; j < 4; ++j) {
                int Nb = (ntb + j) * 16;
                v2f bb = *(const v2f*)&sXP[(kp * NCOL + Nb + ln) * 2];
                acc[j] = wmma_f32(a, bb, acc[j]);
            }
        }
        // ---- epilogue: bias (folded through PA), BN, ReLU, residual, ReLU ----
        #pragma unroll
        for (int j = 0; j < 4; ++j) {
            int n = (ntb + j) * 16 + ln;   // column = t*32 + w
            int t = n >> 5;
            int w = n & 31;
            if (w < V_) {
                #pragma unroll
                for (int i = 0; i < 8; ++i) {
                    int o = Mb + i + 8 * half;
                    float zb = sBias[o]          * sColsum[w]
                             + sBias[O_ + o]     * sColsum[VP + w]
                             + sBias[2 * O_ + o] * sColsum[2 * VP + w];
                    float z = (acc[j][i] + zb) * sScale[o] + sShift[o];
                    z = fmaxf(z, 0.0f);
                    float res = sX[(o * TT + t) * VP + w];  // identity residual
                    float r2 = fmaxf(z + res, 0.0f);
                    size_t go = ((((size_t)n_idx * O_ + o) * T_ + (Tbase + t)) * V_ + w) * M_ + m_idx;
                    out[go] = r2;
                }
            }
        }
    }
}

extern "C" void kernel_launch(void* const* d_in, const int* in_sizes, int n_in,
                              void* d_out, int out_size, void* d_ws, size_t ws_size,
                              hipStream_t stream) {
    const float* x        = (const float*)d_in[0];
    const float* PA       = (const float*)d_in[1];
    const float* conv_w   = (const float*)d_in[2];
    const float* conv_b   = (const float*)d_in[3];
    const float* bn_gamma = (const float*)d_in[4];
    const float* bn_beta  = (const float*)d_in[5];
    const float* bn_mean  = (const float*)d_in[6];
    const float* bn_var   = (const float*)d_in[7];
    float* out = (float*)d_out;

    dim3 grid(T_ / TT, B_);   // 75 x 64 workgroups
    dim3 block(256);          // 8 waves (wave32)
    stgcn_fused_kernel<<<grid, block, 0, stream>>>(
        x, PA, conv_w, conv_b, bn_gamma, bn_beta, bn_mean, bn_var, out);
}